// _Rotation42d_5781025980927
// MI455X (gfx1250) — compile-verified
//
#include <hip/hip_runtime.h>
#include <stdint.h>

// Rotation4-2d: per-sample rotate of (B=256, C=3, H=224, W=224) fp32 by label {0..3}.
// Bandwidth-bound: ~308 MB single-pass traffic -> ~13.2 us floor at 23.3 TB/s.
// CDNA5 paths: double-buffered global->LDS async DMA pipeline (ASYNCcnt depth
// waits) for the transpose labels, conflict-free stride-33 LDS transpose,
// non-temporal 128-bit streams everywhere (no reuse -> don't pollute L2).

#define HDIM 224
#define WDIM 224
#define TILE 32
#define NTILES 7        // 224 / 32 tiles per strip
#define LDS_STRIDE 33   // gcd(33,64)=1 -> conflict-free column gathers

// Native clang vector: required by __builtin_nontemporal_* (HIP's float4 is a
// class type and is rejected).
typedef __attribute__((ext_vector_type(4))) float f32x4;

__device__ __forceinline__ void async_load_f32(unsigned lds_addr,
                                               const float* gaddr) {
  asm volatile("global_load_async_to_lds_b32 %0, %1, off"
               :: "v"(lds_addr), "v"(gaddr) : "memory");
}

__global__ __launch_bounds__(256) void _rot4_2d_kernel(
    const float* __restrict__ x, const int* __restrict__ label,
    float* __restrict__ out) {
  // Double-buffered staging tile: 2 * 32 * 33 * 4 = 8448 B
  __shared__ float tile[2][TILE * LDS_STRIDE];

  const int t = threadIdx.x;
  const int ty = blockIdx.x;        // strip index: output rows [32*ty, 32*ty+32)
  const int plane = blockIdx.y;     // b*3 + c
  const int b = plane / 3;

  const int lab = label[b] & 3;

  const float* __restrict__ src = x   + (size_t)plane * (HDIM * WDIM);
  float*       __restrict__ dst = out + (size_t)plane * (HDIM * WDIM);

  const int r0 = ty * TILE;

  if (lab == 0 || lab == 2) {
    // r0:   out[r][c] = in[r][c]        (identity)
    // r180: out[r][c] = in[H-1-r][c]    (row flip) -- rows contiguous, stream it
    const int r = r0 + (t >> 3);
    const int sr = (lab == 0) ? r : (HDIM - 1 - r);
    const float* srow = src + (size_t)sr * WDIM;
    float*       drow = dst + (size_t)r  * WDIM;
#pragma unroll
    for (int s = 0; s < NTILES; ++s) {
      const int c = s * TILE + ((t & 7) << 2);
      const f32x4 v = __builtin_nontemporal_load((const f32x4*)(srow + c));
      __builtin_nontemporal_store(v, (f32x4*)(drow + c));
    }
    return;
  }

  // r90:  out[i][j] = in[j][i]      -> tile s sources rows [32s, 32s+32)
  // r270: out[i][j] = in[W-1-j][i]  -> tile s sources rows [W-32-32s, W-32s)
  // Stage each 32x32 source tile via async global->LDS DMA; 4 b32 ops per wave
  // per tile. Pipeline depth 2: issue tile s+1, then s_wait_asynccnt 4 drains
  // exactly tile s (async loads complete in order).
  const int oi = t >> 3;   // output row within strip
  const int q  = t & 7;    // float4 column quad within tile

  auto issue = [&](int s, int buf) {
    const int sr0 = (lab == 1) ? (s * TILE) : (WDIM - TILE - s * TILE);
    const float* tsrc = src + (size_t)sr0 * WDIM + r0;
    const unsigned lbase = (unsigned)(uintptr_t)(&tile[buf][0]);
#pragma unroll
    for (int k = 0; k < 4; ++k) {
      const int e = k * 256 + t;     // 0..1023: 1024 floats per tile
      const int r = e >> 5;          // source row within tile
      const int c = e & 31;          // source col within tile (coalesced lanes)
      async_load_f32(lbase + (unsigned)((r * LDS_STRIDE + c) * 4),
                     tsrc + (size_t)r * WDIM + c);
    }
  };

  issue(0, 0);
#pragma unroll
  for (int s = 0; s < NTILES; ++s) {
    if (s + 1 < NTILES) {
      issue(s + 1, (s + 1) & 1);                       // prefetch next tile
      asm volatile("s_wait_asynccnt 0x4" ::: "memory"); // oldest batch landed
    } else {
      asm volatile("s_wait_asynccnt 0x0" ::: "memory");
    }
    __syncthreads();   // all waves' portions of tile s visible in LDS

    const float* lbuf = &tile[s & 1][0];
    f32x4 v;
#pragma unroll
    for (int k = 0; k < 4; ++k) {
      const int jj = 4 * q + k;                          // col within out tile
      const int lr = (lab == 1) ? jj : (TILE - 1 - jj);  // r270: mirrored row
      v[k] = lbuf[lr * LDS_STRIDE + oi];                 // stride-33: no conflicts
    }
    __builtin_nontemporal_store(
        v, (f32x4*)(dst + (size_t)(r0 + oi) * WDIM + (s * TILE + 4 * q)));

    __syncthreads();   // strip reads done before this buffer is DMA-overwritten
  }
}

// Tuple output: append label (int32, bit-copied) after the rotated tensor.
__global__ void _label_tail_kernel(const int* __restrict__ label,
                                   int* __restrict__ tail, int n) {
  const int i = blockIdx.x * blockDim.x + threadIdx.x;
  if (i < n) tail[i] = label[i];
}

extern "C" void kernel_launch(void* const* d_in, const int* in_sizes, int n_in,
                              void* d_out, int out_size, void* d_ws, size_t ws_size,
                              hipStream_t stream) {
  (void)n_in; (void)d_ws; (void)ws_size; (void)out_size;
  const float* x     = (const float*)d_in[0];
  const int*   label = (const int*)d_in[1];
  float*       out   = (float*)d_out;

  const int B = in_sizes[1];   // 256
  const int planes = B * 3;    // 768

  dim3 grid(NTILES, planes);   // (strip row, plane): one div by 3 in the kernel
  _rot4_2d_kernel<<<grid, 256, 0, stream>>>(x, label, out);

  int* tail = (int*)(out + (size_t)planes * HDIM * WDIM);
  _label_tail_kernel<<<(B + 255) / 256, 256, 0, stream>>>(label, tail, B);
}